// GraphCNN_80221399155047
// MI455X (gfx1250) — compile-verified
//
#include <hip/hip_runtime.h>

// ---------------------------------------------------------------------------
// GraphCNN on MI455X (gfx1250).
//
// norm_A is rank-2: norm_A = d d^T - (d.m)(d.m)^T, d = sqrt(deg) (two values:
// sqrt(n) for S nodes, sqrt(n-50) for the 50 "non-S" nodes = {i<64} \ CLASS).
// So norm_A @ Y row i = d[i]*s1 - d[i]*m[i]*s2 with s1 = d^T Y = (d^T x) @ W.T.
// After layer 1 every row is one of TWO vectors (u for S, v for non-S);
// layers 2-3 collapse to 2x256 @ 256x256 matvecs on closed-form recurrences.
//
// Traffic: read x (8MB) + weights (768KB), write out (8MB) -> ~0.7us at
// 23.3 TB/s. Matvecs use V_WMMA_F32_16X16X4_F32 (f32 mandatory: values reach
// ~1e21 by layer 3; f16/bf16 would overflow).
// ---------------------------------------------------------------------------

typedef __attribute__((ext_vector_type(2))) float v2f;
typedef __attribute__((ext_vector_type(4))) float v4f;
typedef __attribute__((ext_vector_type(8))) float v8f;

#define FEAT 256
#define NBLK 64   // partial-reduction blocks (8192/64 = 128 rows each)

// OBJECT_CLASS = {19,20,51..61,63} (14 unique ids) as a 64-bit mask.
__device__ __forceinline__ bool is_non_s(int i) {
    const unsigned long long cls =
        (1ULL << 19) | (1ULL << 20) | (0x7FFULL << 51) | (1ULL << 63);
    return (i < 64) && !((cls >> (i & 63)) & 1ULL);
}

// ---------------------------------------------------------------------------
// Kernel 1: deterministic weighted column sums of x.
//   part[b][c] = sum over block-b rows of w(i)*x[i][c], w = sqrt(n)|sqrt(n-50)
//   r2[c]      = sqrt(n-50) * sum over non-S rows of x[i][c]  (block 0 only)
// ---------------------------------------------------------------------------
__global__ void __launch_bounds__(FEAT)
gcn_reduce_partials(const float* __restrict__ x, float* __restrict__ part,
                    float* __restrict__ r2, int n) {
    const int t = threadIdx.x;            // column
    const int b = blockIdx.x;
    const int rpb = (n + NBLK - 1) / NBLK;
    const int i0 = b * rpb;
    const int i1 = (i0 + rpb < n) ? (i0 + rpb) : n;
    const float sn = sqrtf((float)n);
    const float sm = sqrtf((float)(n - 50));

    float acc = 0.0f;
    for (int i = i0; i < i1; ++i) {
        const float w = is_non_s(i) ? sm : sn;
        acc = fmaf(w, x[(size_t)i * FEAT + t], acc);
    }
    part[b * FEAT + t] = acc;

    if (b == 0) {                          // r2 touches only rows 0..63
        float acc2 = 0.0f;
        const int lim = (n < 64) ? n : 64;
        for (int i = 0; i < lim; ++i) {
            const float sel = is_non_s(i) ? 1.0f : 0.0f;
            acc2 = fmaf(sel, x[(size_t)i * FEAT + t], acc2);
        }
        r2[t] = sm * acc2;
    }
}

// ---------------------------------------------------------------------------
// Kernel 2: finalize r1, then run the 3-layer recurrence with WMMA matvecs.
//
// One workgroup, 512 threads = 16 wave32; wave w owns output columns
// [16w, 16w+16). The zero-padded A (16x256: row0 = state-u, row1 = state-v,
// rows 2..15 = 0) lives in LDS so the A fragment is ONE aligned ds_load_b64
// per k-step -- no per-lane selects, no EXEC divergence in the WMMA loop.
// T = A @ W.T via 64 chained V_WMMA_F32_16X16X4_F32 per wave; result rows
// M=0 / M=1 are c[0]/c[1] in lanes 0..15 (ISA 7.12.2 C/D layout).
// ---------------------------------------------------------------------------
__global__ void __launch_bounds__(512)
gcn_layers_wmma(const float* __restrict__ part, const float* __restrict__ r2g,
                const float* __restrict__ W0, const float* __restrict__ W1,
                const float* __restrict__ W2, float* __restrict__ uvout, int n) {
    __shared__ float Apad[16][FEAT];   // 16 KB; rows 0/1 = state, 2..15 = zero

    const int t = threadIdx.x;

    // Zero padding rows 2..15 (14*256 = 3584 floats, 512 threads -> 7 each).
    for (int idx = t; idx < 14 * FEAT; idx += 512)
        Apad[2 + (idx >> 8)][idx & (FEAT - 1)] = 0.0f;
    // Finalize r1 deterministically into row 0; r2 into row 1.
    if (t < FEAT) {
        float s = 0.0f;
        for (int b = 0; b < NBLK; ++b) s += part[b * FEAT + t];
        Apad[0][t] = s;
        Apad[1][t] = r2g[t];
    }
    __syncthreads();

    const float sn = sqrtf((float)n);
    const float sm = sqrtf((float)(n - 50));
    const float cS = (float)(n - 50) * sn;   // weight of Yu in s1 (layers 2,3)
    const float cN = 50.0f * sm;             // weight of Yv in s1/s2

    const int lane = t & 31;
    const int wave = t >> 5;        // tile id: 16 tiles x 16 cols = 256 cols
    const int m    = lane & 15;     // A row index / B,C column-in-tile
    const int kh   = lane >> 4;     // selects K pair {0,1} vs {2,3}
    const int col  = wave * 16 + m; // global output column (valid lanes 0..15)

    for (int layer = 0; layer < 3; ++layer) {
        const float* __restrict__ W = (layer == 0) ? W0 : ((layer == 1) ? W1 : W2);
        const float* __restrict__ wrow = W + (size_t)col * FEAT + 2 * kh;

        v8f c = {};
        for (int k0 = 0; k0 < FEAT; k0 += 4) {
            const int k = k0 + 2 * kh;
            // A fragment: Apad[m][k..k+1]  (8-byte aligned ds_load_b64)
            const v2f a = *(const v2f*)&Apad[m][k];
            // B fragment: B[kk][nn] = W.T[k][col] = W[col][k] (row-major W)
            const v2f bf = *(const v2f*)(wrow + k0);
            // (neg_a, A, neg_b, B, c_mod, C, reuse_a, reuse_b)
            c = __builtin_amdgcn_wmma_f32_16x16x4_f32(false, a, false, bf,
                                                      (short)0, c, false, false);
        }

        // Tile rows M=0 (t1) and M=1 (t2), valid in lanes 0..15.
        const float t1 = c[0];
        const float t2 = c[1];
        float u, v;
        if (layer == 0) {
            u = fmaxf(sn * t1, 0.0f);               // S-node row
            v = fmaxf(sm * (t1 - t2), 0.0f);        // non-S row
        } else {
            const float s1 = cS * t1 + cN * t2;     // d^T Y
            const float s2 = cN * t2;               // (d.m)^T Y
            u = fmaxf(sn * s1, 0.0f);
            v = fmaxf(sm * (s1 - s2), 0.0f);
        }
        __syncthreads();                 // all waves done reading Apad
        if (lane < 16) { Apad[0][col] = u; Apad[1][col] = v; }
        __syncthreads();                 // new state visible to all waves
    }

    if (lane < 16) {
        uvout[col]        = Apad[0][col];   // final u (S rows)
        uvout[FEAT + col] = Apad[1][col];   // final v (non-S rows)
    }
}

// ---------------------------------------------------------------------------
// Kernel 3: broadcast the two final rows to the [n, 256] output, float4 stores.
// ---------------------------------------------------------------------------
__global__ void __launch_bounds__(256)
gcn_broadcast(const float* __restrict__ uv, float* __restrict__ out, int n) {
    const int g = blockIdx.x * blockDim.x + threadIdx.x;   // float4 index
    const int per_row = FEAT / 4;
    if (g >= n * per_row) return;
    const int row = g / per_row;
    const int c   = (g - row * per_row) * 4;
    const float* src = is_non_s(row) ? (uv + FEAT) : uv;
    const v4f val = *(const v4f*)(src + c);
    *(v4f*)(out + (size_t)row * FEAT + c) = val;
}

extern "C" void kernel_launch(void* const* d_in, const int* in_sizes, int n_in,
                              void* d_out, int out_size, void* d_ws, size_t ws_size,
                              hipStream_t stream) {
    const float* x  = (const float*)d_in[0];
    const float* W0 = (const float*)d_in[1];
    const float* W1 = (const float*)d_in[2];
    const float* W2 = (const float*)d_in[3];
    const int n = in_sizes[0] / FEAT;   // 8192

    float* ws   = (float*)d_ws;
    float* part = ws;                   // NBLK*FEAT floats
    float* r2   = part + NBLK * FEAT;   // FEAT floats
    float* uv   = r2 + FEAT;            // 2*FEAT floats  (~68 KB total)

    gcn_reduce_partials<<<NBLK, FEAT, 0, stream>>>(x, part, r2, n);
    gcn_layers_wmma<<<1, 512, 0, stream>>>(part, r2, W0, W1, W2, uv, n);

    const int nvec = n * (FEAT / 4);
    gcn_broadcast<<<(nvec + 255) / 256, 256, 0, stream>>>(uv, (float*)d_out, n);
}